// GCNGRU_38920993636863
// MI455X (gfx1250) — compile-verified
//
#include <hip/hip_runtime.h>
#include <cstdint>

typedef __attribute__((ext_vector_type(2))) float v2f;
typedef __attribute__((ext_vector_type(8))) float v8f;
typedef __attribute__((ext_vector_type(2))) int   v2i;
typedef __attribute__((address_space(1))) v2i* gas_v2i_ptr;  // global (AS1)
typedef __attribute__((address_space(3))) v2i* las_v2i_ptr;  // LDS (AS3)

#define TB 256
#define WPB 8  // waves per block (wave32)

#if defined(__HIP_DEVICE_COMPILE__) && __has_builtin(__builtin_amdgcn_global_load_async_to_lds_b64)
#define USE_ASYNC_LDS 1
#else
#define USE_ASYNC_LDS 0
#endif

// ---------------- setup kernels ----------------
__global__ void k_zero_u32(unsigned* __restrict__ p, int n) {
  int i = blockIdx.x * blockDim.x + threadIdx.x;
  if (i < n) p[i] = 0u;
}
__global__ void k_zero_f32(float* __restrict__ p, int n) {
  int i = blockIdx.x * blockDim.x + threadIdx.x;
  if (i < n) p[i] = 0.0f;
}
__global__ void k_deg(const int* __restrict__ ei, unsigned* __restrict__ deg, int E) {
  int e = blockIdx.x * blockDim.x + threadIdx.x;
  if (e < E) atomicAdd(&deg[ei[E + e]], 1u);
}
__global__ void k_dinv(const unsigned* __restrict__ deg, float* __restrict__ dinv, int n) {
  int i = blockIdx.x * blockDim.x + threadIdx.x;
  if (i < n) dinv[i] = rsqrtf((float)deg[i] + 1.0f);
}
__global__ void k_norm(const int* __restrict__ ei, const float* __restrict__ dinv,
                       float* __restrict__ nrm, int E) {
  int e = blockIdx.x * blockDim.x + threadIdx.x;
  if (e < E) nrm[e] = dinv[ei[e]] * dinv[ei[E + e]];
}

// ---------------- GEMM: XW = X(N x 16) @ Wg(16 x 64), f32 WMMA ----------------
// Wg row-major (K x 64). Each wave computes a 16-row x 64-col strip. Wg is 4KB -> L2/L1 hot.
__global__ void k_gemm_x_wgcn(const float* __restrict__ X, const float* __restrict__ Wg,
                              float* __restrict__ XW, int rowTiles) {
  int wave = (blockIdx.x * blockDim.x + threadIdx.x) >> 5;
  int lane = threadIdx.x & 31;
  if (wave >= rowTiles) return;              // wave-uniform guard (EXEC stays full)
  int row0 = wave << 4;
  int m = lane & 15;
  int khalf = (lane >> 4) << 1;              // 0 or 2
  int nn = lane & 15;
  v8f acc0 = {}, acc1 = {}, acc2 = {}, acc3 = {};
  const float* xrow = X + (size_t)(row0 + m) * 16;
#pragma unroll
  for (int k0 = 0; k0 < 16; k0 += 4) {
    int k = k0 + khalf;
    v2f a = *(const v2f*)(xrow + k);
    v2f b0, b1, b2, b3;
    b0.x = Wg[k * 64 +  0 + nn]; b0.y = Wg[(k + 1) * 64 +  0 + nn];
    b1.x = Wg[k * 64 + 16 + nn]; b1.y = Wg[(k + 1) * 64 + 16 + nn];
    b2.x = Wg[k * 64 + 32 + nn]; b2.y = Wg[(k + 1) * 64 + 32 + nn];
    b3.x = Wg[k * 64 + 48 + nn]; b3.y = Wg[(k + 1) * 64 + 48 + nn];
    acc0 = __builtin_amdgcn_wmma_f32_16x16x4_f32(false, a, false, b0, (short)0, acc0, false, false);
    acc1 = __builtin_amdgcn_wmma_f32_16x16x4_f32(false, a, false, b1, (short)0, acc1, false, false);
    acc2 = __builtin_amdgcn_wmma_f32_16x16x4_f32(false, a, false, b2, (short)0, acc2, false, false);
    acc3 = __builtin_amdgcn_wmma_f32_16x16x4_f32(false, a, false, b3, (short)0, acc3, false, false);
  }
  int mbase = row0 + ((lane >> 4) << 3);
#pragma unroll
  for (int v = 0; v < 8; ++v) {
    size_t r = (size_t)(mbase + v) * 64;
    XW[r +  0 + nn] = acc0[v];
    XW[r + 16 + nn] = acc1[v];
    XW[r + 32 + nn] = acc2[v];
    XW[r + 48 + nn] = acc3[v];
  }
}

// ---------------- GEMM: C = A(N x 64) @ W^T + bias, W stored (192 x 64) row-major ----------------
// W (48KB) is staged once per workgroup into LDS (transposed, padded pitch) via the
// gfx1250 async global->LDS path; each wave then computes a full 16-row x 192-col strip
// (12 WMMA accumulators), so A is read exactly once.
#define LDS_PITCH 193   // float2 pitch: (193*2) % 64 != 0 -> spreads DS banks
__global__ void __launch_bounds__(TB) k_gemm_aw_t(
    const float* __restrict__ A, const float* __restrict__ W,
    const float* __restrict__ bias, float* __restrict__ C, int rowTiles) {
  __shared__ v2f ldsW[32 * LDS_PITCH];  // [kpair][oc], kpair=k/2 in 0..31, oc in 0..191

  // ---- stage W transposed: ldsW[kp*PITCH + o] = {W[o*64+2kp], W[o*64+2kp+1]} ----
  for (int idx = threadIdx.x; idx < 192 * 32; idx += TB) {
    int o  = idx >> 5;       // 0..191  (32 consecutive tids cover one 256B W row)
    int kp = idx & 31;       // 0..31
    const float* src = W + (size_t)o * 64 + (kp << 1);
    v2f* dst = &ldsW[kp * LDS_PITCH + o];
#if USE_ASYNC_LDS
    __builtin_amdgcn_global_load_async_to_lds_b64(
        (gas_v2i_ptr)src, (las_v2i_ptr)dst, 0, 0);
#else
    *dst = *(const v2f*)src;
#endif
  }
#if USE_ASYNC_LDS
#if __has_builtin(__builtin_amdgcn_s_wait_asynccnt)
  __builtin_amdgcn_s_wait_asynccnt(0);
#else
  asm volatile("s_wait_asynccnt 0x0" ::: "memory");
#endif
#endif
  __syncthreads();

  int wave = (blockIdx.x * blockDim.x + threadIdx.x) >> 5;
  int lane = threadIdx.x & 31;
  if (wave >= rowTiles) return;              // wave-uniform; after barrier
  int row0 = wave << 4;
  int m = lane & 15;
  int kplane = lane >> 4;                    // 0 or 1 (K pair select)
  int nn = lane & 15;

  v8f acc[12];
#pragma unroll
  for (int c = 0; c < 12; ++c) acc[c] = (v8f){};

  const float* arow = A + (size_t)(row0 + m) * 64;
#pragma unroll 1
  for (int k0 = 0; k0 < 64; k0 += 4) {
    int kpb = (k0 >> 1) + kplane;
    v2f a = *(const v2f*)(arow + k0 + (kplane << 1));
    const v2f* brow = &ldsW[kpb * LDS_PITCH + nn];
#pragma unroll
    for (int c = 0; c < 12; ++c) {
      v2f b = brow[c * 16];
      acc[c] = __builtin_amdgcn_wmma_f32_16x16x4_f32(false, a, false, b, (short)0, acc[c],
                                                     false, false);
    }
  }

  float bb[12];
#pragma unroll
  for (int c = 0; c < 12; ++c) bb[c] = bias[c * 16 + nn];
  int mbase = row0 + ((lane >> 4) << 3);
#pragma unroll
  for (int v = 0; v < 8; ++v) {
    size_t r = (size_t)(mbase + v) * 192;
#pragma unroll
    for (int c = 0; c < 12; ++c) C[r + c * 16 + nn] = acc[c][v] + bb[c];
  }
}

// ---------------- agg = self_norm * xw + b_gcn ----------------
__global__ void k_agg_init(const float* __restrict__ xw, const float* __restrict__ dinv,
                           const float* __restrict__ bgcn, float* __restrict__ agg, int N) {
  int tid = blockIdx.x * blockDim.x + threadIdx.x;
  if (tid >= N * 16) return;
  int n = tid >> 4;
  int j = (tid & 15) << 2;
  float di = dinv[n];
  float sn = di * di;
  float4 x = *(const float4*)(xw + (size_t)n * 64 + j);
  float4 b = *(const float4*)(bgcn + j);
  float4 r;
  r.x = sn * x.x + b.x; r.y = sn * x.y + b.y;
  r.z = sn * x.z + b.z; r.w = sn * x.w + b.w;
  *(float4*)(agg + (size_t)n * 64 + j) = r;
}

// ---------------- edge scatter: agg[dst] += norm[e] * xw[src] ----------------
__global__ void k_scatter(const int* __restrict__ ei, const float* __restrict__ nrm,
                          const float* __restrict__ xw, float* __restrict__ agg, int E) {
  int tid = blockIdx.x * blockDim.x + threadIdx.x;
  if (tid >= E * 16) return;
  int e = tid >> 4;
  int j = (tid & 15) << 2;
  int s = ei[e];
  int d = ei[E + e];
  float w = nrm[e];
  float4 x = *(const float4*)(xw + (size_t)s * 64 + j);
  float* a = agg + (size_t)d * 64 + j;
  unsafeAtomicAdd(a + 0, x.x * w);
  unsafeAtomicAdd(a + 1, x.y * w);
  unsafeAtomicAdd(a + 2, x.z * w);
  unsafeAtomicAdd(a + 3, x.w * w);
}

// ---------------- GRU gates: h = (1-z)*tanh(xn + r*hn) + z*h ----------------
__global__ void k_gates(const float* __restrict__ gx, const float* __restrict__ gh,
                        float* __restrict__ h, int N) {
  int tid = blockIdx.x * blockDim.x + threadIdx.x;
  if (tid >= N * 64) return;
  int n = tid >> 6;
  int j = tid & 63;
  size_t g = (size_t)n * 192 + j;
  float r = 1.0f / (1.0f + __expf(-(gx[g] + gh[g])));
  float z = 1.0f / (1.0f + __expf(-(gx[g + 64] + gh[g + 64])));
  float nc = tanhf(gx[g + 128] + r * gh[g + 128]);
  size_t hi = (size_t)n * 64 + j;
  float hp = h[hi];
  h[hi] = (1.0f - z) * nc + z * hp;
}

// ---------------- output head: out[p*N + n] = h[n] . W_out[p] + b_out[p] ----------------
__global__ void k_head(const float* __restrict__ h, const float* __restrict__ Wo,
                       const float* __restrict__ bo, float* __restrict__ out, int N) {
  int n = blockIdx.x * blockDim.x + threadIdx.x;
  int p = blockIdx.y;
  if (n >= N) return;
  const float* hr = h + (size_t)n * 64;
  const float* wr = Wo + (size_t)p * 64;
  float acc = bo[p];
#pragma unroll
  for (int j = 0; j < 64; j += 4) {
    float4 hv = *(const float4*)(hr + j);
    float4 wv = *(const float4*)(wr + j);
    acc += hv.x * wv.x + hv.y * wv.y + hv.z * wv.z + hv.w * wv.w;
  }
  out[(size_t)p * N + n] = acc;
}

extern "C" void kernel_launch(void* const* d_in, const int* in_sizes, int n_in,
                              void* d_out, int out_size, void* d_ws, size_t ws_size,
                              hipStream_t stream) {
  const float* x_seq = (const float*)d_in[0];
  const int*   ei    = (const int*)d_in[1];
  // d_in[2] = edge_weight: unused (faithful to reference)
  const float* W_gcn = (const float*)d_in[3];
  const float* b_gcn = (const float*)d_in[4];
  const float* W_ih  = (const float*)d_in[5];
  const float* W_hh  = (const float*)d_in[6];
  const float* b_ih  = (const float*)d_in[7];
  const float* b_hh  = (const float*)d_in[8];
  const float* W_out = (const float*)d_in[9];
  const float* b_out = (const float*)d_in[10];

  const int T = 24, F = 16, H = 64, G = 192;
  const int E = in_sizes[1] / 2;
  const int N = in_sizes[0] / (T * F);
  const int P = out_size / N;

  char* ws = (char*)d_ws;
  size_t off = 0;
  auto carve = [&](size_t bytes) -> void* {
    void* p = ws + off;
    off += (bytes + 255) & ~(size_t)255;
    return p;
  };
  unsigned* deg  = (unsigned*)carve((size_t)N * 4);
  float* dinv    = (float*)carve((size_t)N * 4);
  float* nrm     = (float*)carve((size_t)E * 4);
  float* xw      = (float*)carve((size_t)N * H * 4);
  float* agg     = (float*)carve((size_t)N * H * 4);
  float* gx      = (float*)carve((size_t)N * G * 4);
  float* gh      = (float*)carve((size_t)N * G * 4);
  float* h       = (float*)carve((size_t)N * H * 4);
  (void)ws_size; (void)n_in;

  auto cdiv = [](int a, int b) { return (a + b - 1) / b; };

  // Graph setup (timestep-invariant)
  k_zero_u32<<<cdiv(N, 256), 256, 0, stream>>>(deg, N);
  k_deg<<<cdiv(E, 256), 256, 0, stream>>>(ei, deg, E);
  k_dinv<<<cdiv(N, 256), 256, 0, stream>>>(deg, dinv, N);
  k_norm<<<cdiv(E, 256), 256, 0, stream>>>(ei, dinv, nrm, E);
  k_zero_f32<<<cdiv(N * H, 256), 256, 0, stream>>>(h, N * H);

  const int rowTiles = N / 16;                       // N = 50000 -> 3125 exact
  const int xwBlocks = cdiv(rowTiles, WPB);
  const int gBlocks  = cdiv(rowTiles, WPB);          // one wave per 16x192 strip

  for (int t = 0; t < T; ++t) {
    const float* xt = x_seq + (size_t)t * N * F;
    k_gemm_x_wgcn<<<xwBlocks, TB, 0, stream>>>(xt, W_gcn, xw, rowTiles);
    k_agg_init<<<cdiv(N * 16, 256), 256, 0, stream>>>(xw, dinv, b_gcn, agg, N);
    k_scatter<<<cdiv(E * 16, 256), 256, 0, stream>>>(ei, nrm, xw, agg, E);
    k_gemm_aw_t<<<gBlocks, TB, 0, stream>>>(agg, W_ih, b_ih, gx, rowTiles);
    k_gemm_aw_t<<<gBlocks, TB, 0, stream>>>(h,   W_hh, b_hh, gh, rowTiles);
    k_gates<<<cdiv(N * H, 256), 256, 0, stream>>>(gx, gh, h, N);
  }

  dim3 hgrid(cdiv(N, 256), P);
  k_head<<<hgrid, 256, 0, stream>>>(h, W_out, b_out, (float*)d_out, N);
}